// WaveletEncode1d_82798379532448
// MI455X (gfx1250) — compile-verified
//
#include <hip/hip_runtime.h>

// 6-level 1D wavelet decomposition, (16, 2, 1<<20) f32, two 9-tap filters.
//
// Dataflow (≈530 MB total vs reference ≈1.6 GB):
//  - Per level j: stride-2 dual-filter conv of low band a_{j-1} via
//    V_WMMA_F32_16X16X4_F32, filters in A (M dim), data in B (N dim):
//      D[f][t] = sum_k filt[f][k] * padded[2t+k],  K=9 padded to 12.
//    D row 0 = low band, D row 1 = high band, N = 16 consecutive t across
//    lanes 0..15 -> coalesced 16-lane extraction stores.
//  - Low band a_j -> ping-pong in d_ws (level 6 -> d_out ch 0..1).
//  - Detail d_j written directly to final permuted position:
//      block base = 2^(m+1), m = 6-j
//      c_f = ch | (bitrev_m(t & (2^m-1)) << 1)  (via __brev),  t_f = t >> m
//  - Filter operand hoisted per wave; 8 t-blocks per wave; reflect padding
//    only on the two edge waves of each row.
//
// ws peak: a1 (64 MiB) + a2 (32 MiB) ping-pong = ~100.7 MB.

typedef __attribute__((ext_vector_type(2))) float v2f;
typedef __attribute__((ext_vector_type(8))) float v8f;

__device__ __forceinline__ int reflect_idx(int q, int L) {
    q = (q < 0) ? -q : q;                    // reflect (no edge repeat), pad = 4
    q = (q >= L) ? (2 * L - 2 - q) : q;
    return q;
}

#define KITER 8  // 16-output t-blocks per wave

__global__ __launch_bounds__(256) void wavelet_level_wmma(
    const float* __restrict__ in,    // (16, 2, Lin) contiguous
    const float* __restrict__ filt,  // (2, 9) reversed filters (as given)
    float* __restrict__ aout,        // low band destination
    float* __restrict__ dout,        // final output base (16, 128, 16384)
    int Lin, int m,
    long long aBatch, long long aCh)
{
    const int lane  = threadIdx.x & 31;
    const int wid   = threadIdx.x >> 5;
    const int rowId = blockIdx.y;             // 0..31 = n*2 + ch
    const int n  = rowId >> 1;
    const int ch = rowId & 1;
    const int Lout  = Lin >> 1;
    const int span0 = (blockIdx.x * 8 + wid) * (16 * KITER);  // first t

    const float* __restrict__ row = in + (size_t)rowId * (size_t)Lin;

    const int nn   = lane & 15;               // N index (t within block) for B
    const int koff = (lane >> 4) << 1;        // K sub-index 0 or 2

    // ---- A operand (filters), loop-invariant. A[m=f][k]: lane f=nn,
    //      k = koff+r within chunk; tap = 4c+koff+r; rows f>=2 and taps>=9 zero.
    v2f Af0, Af1, Af2;
    {
        const int  ffc = (nn < 2) ? nn : 1;   // clamp gather index (OOB-safe)
        const bool fe  = (nn < 2);
        int t0a = koff, t1a = 4 + koff, t2a = 8 + koff;
        float v00 = filt[ffc * 9 + t0a];
        float v01 = filt[ffc * 9 + t0a + 1];
        float v10 = filt[ffc * 9 + t1a];
        float v11 = filt[ffc * 9 + t1a + 1];
        float v20 = filt[ffc * 9 + ((t2a     < 9) ? t2a     : 8)];
        float v21 = filt[ffc * 9 + ((t2a + 1 < 9) ? t2a + 1 : 8)];
        Af0 = (v2f){ fe ? v00 : 0.0f, fe ? v01 : 0.0f };
        Af1 = (v2f){ fe ? v10 : 0.0f, fe ? v11 : 0.0f };
        Af2 = (v2f){ (fe && t2a     < 9) ? v20 : 0.0f,
                     (fe && t2a + 1 < 9) ? v21 : 0.0f };
    }

    float* __restrict__ lowp = aout + (size_t)n * (size_t)aBatch
                                    + (size_t)ch * (size_t)aCh;
    float* __restrict__ bout = dout + ((size_t)n << 21);   // 128*16384 per batch
    const int cbase = 1 << (m + 1);
    const int mask  = (1 << m) - 1;

    // reflect needed only where 2t+tap-4 leaves [0, Lin): first & last span
    const bool edge = (span0 == 0) || (span0 + 16 * KITER == Lout);

    #pragma unroll 2
    for (int it = 0; it < KITER; ++it) {
        const int t0   = span0 + it * 16;
        const int base = 2 * t0 + 2 * nn + koff - 4;   // even -> 8B aligned

        // ---- B operand (data): B[kk][t]: lane nn holds padded[2t+4c+koff+r]
        v2f B0, B1, B2;
        if (!edge) {
            B0 = *(const v2f*)(row + base);
            B1 = *(const v2f*)(row + base + 4);
            B2 = *(const v2f*)(row + base + 8);
        } else {
            B0 = (v2f){ row[reflect_idx(base + 0, Lin)],
                        row[reflect_idx(base + 1, Lin)] };
            B1 = (v2f){ row[reflect_idx(base + 4, Lin)],
                        row[reflect_idx(base + 5, Lin)] };
            B2 = (v2f){ row[reflect_idx(base + 8, Lin)],
                        row[reflect_idx(base + 9, Lin)] };
        }

        v8f acc = {0.f, 0.f, 0.f, 0.f, 0.f, 0.f, 0.f, 0.f};
        acc = __builtin_amdgcn_wmma_f32_16x16x4_f32(false, Af0, false, B0,
                                                    (short)0, acc, false, false);
        acc = __builtin_amdgcn_wmma_f32_16x16x4_f32(false, Af1, false, B1,
                                                    (short)0, acc, false, false);
        acc = __builtin_amdgcn_wmma_f32_16x16x4_f32(false, Af2, false, B2,
                                                    (short)0, acc, false, false);

        // ---- D: VGPR0 lanes 0-15 = low[t0+lane], VGPR1 lanes 0-15 = high ----
        if (lane < 16) {
            const int t = t0 + lane;
            lowp[t] = acc[0];
            const int tf = t >> m;
            const unsigned rev =
                m ? (__brev((unsigned)(t & mask)) >> (32 - m)) : 0u;
            const int cf = ch | ((int)rev << 1);
            bout[(size_t)(cbase + cf) * 16384 + (size_t)tf] = acc[1];
        }
    }
}

extern "C" void kernel_launch(void* const* d_in, const int* in_sizes, int n_in,
                              void* d_out, int out_size, void* d_ws, size_t ws_size,
                              hipStream_t stream) {
    (void)in_sizes; (void)n_in; (void)out_size; (void)ws_size;
    const float* x    = (const float*)d_in[0];   // (16, 2, 1<<20) f32
    const float* filt = (const float*)d_in[1];   // (2, 9) f32
    // d_in[2] = levels (always 6 per setup_inputs)
    float* out = (float*)d_out;                  // (16, 128, 16384) f32
    float* ws  = (float*)d_ws;

    const int L0 = 1 << 20;
    const size_t a1elems = (size_t)16 * 2 * (L0 / 2);   // 16,777,216 floats
    float* bufA = ws;                  // a1, a3, a5
    float* bufB = ws + a1elems;        // a2, a4

    const float* cur = x;
    for (int j = 1; j <= 6; ++j) {
        const int Lin  = L0 >> (j - 1);
        const int Lout = Lin >> 1;
        const int m    = 6 - j;
        float* aout;
        long long aB, aC;
        if (j == 6) { aout = out; aB = 1 << 21; aC = 16384; }
        else        { aout = (j & 1) ? bufA : bufB; aB = 2LL * Lout; aC = Lout; }
        const int wavesPerRow = Lout / (16 * KITER);   // >= 128, power of two
        dim3 grid(wavesPerRow / 8, 32);                // 8 waves / 256-thr block
        wavelet_level_wmma<<<grid, 256, 0, stream>>>(cur, filt, aout, out,
                                                     Lin, m, aB, aC);
        cur = aout;
    }
}